// SAGE_24773371363586
// MI455X (gfx1250) — compile-verified
//
#include <hip/hip_runtime.h>

typedef __attribute__((ext_vector_type(2))) float v2f;
typedef __attribute__((ext_vector_type(8))) float v8f;

#define D_IN 128

// ---------------------------------------------------------------------------
// Hardware float atomic add (global_atomic_add_f32, not a CAS loop).
// ---------------------------------------------------------------------------
__device__ __forceinline__ void atomic_add_f32(float* p, float v) {
  unsafeAtomicAdd(p, v);
}

// ---------------------------------------------------------------------------
// Layer init: agg[i] = X[i] (self-loop contribution), deg[i] = 1.0 (self loop).
// One wave per row, float4 per lane -> 512B fully-coalesced per row.
// ---------------------------------------------------------------------------
__global__ void init_layer_kernel(const float* __restrict__ X,
                                  float* __restrict__ agg,
                                  float* __restrict__ deg, int n) {
  int t = blockIdx.x * blockDim.x + threadIdx.x;
  int row = t >> 5;          // 32 lanes cover 128 floats as float4
  int c4  = (t & 31) << 2;
  if (row >= n) return;
  float4 v = *(const float4*)(X + (size_t)row * D_IN + c4);
  *(float4*)(agg + (size_t)row * D_IN + c4) = v;
  if (deg != nullptr && c4 == 0) deg[row] = 1.0f;
}

// ---------------------------------------------------------------------------
// Edge scatter: agg[dst] += X[src] (mean-agg numerator), deg[dst] += 1.
// One wave per edge: coalesced 512B gather of the source row, 4 f32 atomics
// per lane into the destination row.
// ---------------------------------------------------------------------------
__global__ void edge_scatter_kernel(const float* __restrict__ X,
                                    const int* __restrict__ src,
                                    const int* __restrict__ dst,
                                    float* __restrict__ agg,
                                    float* __restrict__ deg, int ne) {
  int t = blockIdx.x * blockDim.x + threadIdx.x;
  int e  = t >> 5;
  int c4 = (t & 31) << 2;
  if (e >= ne) return;
  int s = src[e];
  int d = dst[e];
  float4 v = *(const float4*)(X + (size_t)s * D_IN + c4);
  float* ap = agg + (size_t)d * D_IN + c4;
  atomic_add_f32(ap + 0, v.x);
  atomic_add_f32(ap + 1, v.y);
  atomic_add_f32(ap + 2, v.z);
  atomic_add_f32(ap + 3, v.w);
  if (deg != nullptr && c4 == 0) atomic_add_f32(&deg[d], 1.0f);
}

// ---------------------------------------------------------------------------
// Fused SAGE GEMM: out = [Aself | (Agg/deg)] @ [Wself ; Wneigh] + b (opt ReLU)
// fp32 WMMA 16x16x4. Block covers 16 rows x DOUT cols; one wave per 16x16
// output tile (DOUT/16 waves per block). A tile staged in LDS with deg
// scaling, padded stride (260 floats) so the 16 M-lanes hit distinct banks.
//
// Fragment layouts per CDNA5 ISA 7.12.2:
//   A (16x4 f32):  lanes 0-15 hold M=lane with K={k0,k0+1}; lanes 16-31 hold
//                  M=lane-16 with K={k0+2,k0+3}  -> koff = (lane>>4)*2
//   B (4x16 f32):  symmetric: lanes hold N=lane&15, K pair selected by koff
//   C (16x16 f32): VGPR r -> M = r + (lane>>4)*8, N = lane&15
// ---------------------------------------------------------------------------
template <int DOUT, int RELU>
__global__ __launch_bounds__(DOUT * 2) void sage_gemm_kernel(
    const float* __restrict__ Aself,   // [n,128]
    const float* __restrict__ Agg,     // [n,128]
    const float* __restrict__ deg,     // [n]
    const float* __restrict__ Wself,   // [128,DOUT] row-major
    const float* __restrict__ Wneigh,  // [128,DOUT] row-major
    const float* __restrict__ bias,    // [DOUT]
    float* __restrict__ out,           // [n,DOUT]
    int n) {
  constexpr int LDA = 2 * D_IN + 4;  // 260: m*260 % 64 distinct for m=0..15
  __shared__ float As[16 * LDA];

  const int tid  = threadIdx.x;
  const int row0 = blockIdx.x * 16;

  // Cooperative stage of the stacked A tile: cols [0,128)=self,
  // [128,256)=agg/deg. Consecutive tids -> consecutive columns (coalesced).
  for (int idx = tid; idx < 16 * D_IN; idx += DOUT * 2) {
    int r = idx >> 7;
    int c = idx & (D_IN - 1);
    int g = row0 + r;
    float a = 0.0f, hn = 0.0f;
    if (g < n) {
      a  = Aself[(size_t)g * D_IN + c];
      hn = Agg[(size_t)g * D_IN + c] * (1.0f / deg[g]);
    }
    As[r * LDA + c]        = a;
    As[r * LDA + D_IN + c] = hn;
  }
  __syncthreads();

  const int lane = tid & 31;
  const int wave = tid >> 5;                  // column tile
  const int m    = lane & 15;
  const int koff = (lane >> 4) << 1;          // 0 or 2
  const int col  = (wave << 4) + (lane & 15); // global output column

  v8f acc = {};
  const float* Asrow = &As[m * LDA];

  #pragma unroll
  for (int phase = 0; phase < 2; ++phase) {
    const float* W  = phase ? Wneigh : Wself;
    const float* Ap = Asrow + phase * D_IN;
    #pragma unroll 4
    for (int k0 = 0; k0 < D_IN; k0 += 4) {
      v2f a = *(const v2f*)(Ap + k0 + koff);  // 8B-aligned LDS read (ds_load_b64)
      v2f b;
      b.x = W[(size_t)(k0 + koff + 0) * DOUT + col];
      b.y = W[(size_t)(k0 + koff + 1) * DOUT + col];
      // D = A*B + C ; emits v_wmma_f32_16x16x4_f32
      acc = __builtin_amdgcn_wmma_f32_16x16x4_f32(
          /*neg_a=*/false, a, /*neg_b=*/false, b,
          /*c_mod=*/(short)0, acc, /*reuse_a=*/false, /*reuse_b=*/false);
    }
  }

  const float bv = bias[col];
  const int rbase = row0 + ((lane >> 4) << 3);
  #pragma unroll
  for (int r = 0; r < 8; ++r) {
    int g = rbase + r;
    if (g < n) {
      float v = acc[r] + bv;
      if (RELU) v = fmaxf(v, 0.0f);
      out[(size_t)g * DOUT + col] = v;
    }
  }
}

// ---------------------------------------------------------------------------
// Launch: layer1 (d=128, relu) then layer2 (d=64). deg computed once, reused.
// Workspace: agg[n*128] | h[n*128] | deg[n]  (~103 MB), fully written per call.
// ---------------------------------------------------------------------------
extern "C" void kernel_launch(void* const* d_in, const int* in_sizes, int n_in,
                              void* d_out, int out_size, void* d_ws, size_t ws_size,
                              hipStream_t stream) {
  const float* x       = (const float*)d_in[0];
  const int*   src     = (const int*)d_in[1];
  const int*   dst     = (const int*)d_in[2];
  const float* Wself1  = (const float*)d_in[3];
  const float* Wneigh1 = (const float*)d_in[4];
  const float* b1      = (const float*)d_in[5];
  const float* Wself2  = (const float*)d_in[6];
  const float* Wneigh2 = (const float*)d_in[7];
  const float* b2      = (const float*)d_in[8];
  float* out = (float*)d_out;

  const int n  = in_sizes[0] / D_IN;   // 100000
  const int ne = in_sizes[1];          // 1000000

  float* agg = (float*)d_ws;
  float* h   = agg + (size_t)n * D_IN;
  float* deg = h + (size_t)n * D_IN;

  const dim3 bs(256);
  const dim3 gNode((n * 32 + 255) / 256);   // one wave per row
  const dim3 gEdge((ne * 32 + 255) / 256);  // one wave per edge
  const dim3 gGemm((n + 15) / 16);

  // ---- Layer 1: h = relu(x@Ws1 + mean_agg(x)@Wn1 + b1) ----
  init_layer_kernel<<<gNode, bs, 0, stream>>>(x, agg, deg, n);
  edge_scatter_kernel<<<gEdge, bs, 0, stream>>>(x, src, dst, agg, deg, ne);
  sage_gemm_kernel<128, 1><<<gGemm, dim3(256), 0, stream>>>(
      x, agg, deg, Wself1, Wneigh1, b1, h, n);

  // ---- Layer 2: out = h@Ws2 + mean_agg(h)@Wn2 + b2 (deg unchanged) ----
  init_layer_kernel<<<gNode, bs, 0, stream>>>(h, agg, nullptr, n);
  edge_scatter_kernel<<<gEdge, bs, 0, stream>>>(h, src, dst, agg, nullptr, ne);
  sage_gemm_kernel<64, 0><<<gGemm, dim3(128), 0, stream>>>(
      h, agg, deg, Wself2, Wneigh2, b2, out, n);
}